// AppearanceComposability_1795296330400
// MI455X (gfx1250) — compile-verified
//
#include <hip/hip_runtime.h>
#include <cstdint>

#define TPB        256
#define HW         4096        // 64*64 floats per (b,c) channel
#define OUTF_PER_CH 36864      // 9 * 4096 output floats per channel
#define OUTV_PER_CH 9216       // /4 -> 128-bit stores per channel

typedef float v4f __attribute__((ext_vector_type(4)));

__global__ __launch_bounds__(TPB)
void ac_mul_kernel(const float* __restrict__ key,
                   const float* __restrict__ qry,
                   float* __restrict__ out)
{
    __shared__ float lds_key[HW];   // 16 KB
    __shared__ float lds_qry[HW];   // 16 KB

    const int tid = threadIdx.x;
    const int ch  = blockIdx.x;                         // b*256 + c
    const float* kg = key + (size_t)ch * HW;
    const float* qg = qry + (size_t)ch * HW;

    // Wave-relative LDS byte addresses: low 32 bits of the generic (flat)
    // address of a __shared__ object equal the LDS byte offset used by
    // DS / async-to-LDS instructions.
    const uint32_t kbase = (uint32_t)(uintptr_t)(&lds_key[0]);
    const uint32_t qbase = (uint32_t)(uintptr_t)(&lds_qry[0]);

    // Stage this channel's key+query tiles into LDS with CDNA5 async copies.
    // 1024 float4 per tile / 256 threads = 4 issues per tile per thread.
#pragma unroll
    for (int it = 0; it < 4; ++it) {
        const int i = tid + it * TPB;                   // float4 index
        const uint64_t ka = (uint64_t)(uintptr_t)(kg + i * 4);
        const uint64_t qa = (uint64_t)(uintptr_t)(qg + i * 4);
        const uint32_t kl = kbase + (uint32_t)(i * 16);
        const uint32_t ql = qbase + (uint32_t)(i * 16);
        asm volatile("global_load_async_to_lds_b128 %0, %1, off"
                     :: "v"(kl), "v"(ka) : "memory");
        asm volatile("global_load_async_to_lds_b128 %0, %1, off"
                     :: "v"(ql), "v"(qa) : "memory");
    }
    asm volatile("s_wait_asynccnt 0" ::: "memory");
    __syncthreads();

    v4f* outv = (v4f*)out + (size_t)ch * OUTV_PER_CH;

    // out[f] = F(key,f) * F(query, (f/9)*9 + 4)   (raw-reshape interleave)
    for (int v = tid; v < OUTV_PER_CH; v += TPB) {
        const int f0 = v << 2;
        v4f rv;
#pragma unroll
        for (int j = 0; j < 4; ++j) {
            const int fk = f0 + j;
            // ---- key factor ----
            const int p  = fk >> 12;                    // patch idx 0..8
            const int l  = fk & 4095;
            const int hh = (l >> 6) + p / 3 - 1;
            const int ww = (l & 63) + p % 3 - 1;
            const float kv = ((unsigned)hh < 64u && (unsigned)ww < 64u)
                                 ? lds_key[(hh << 6) + ww] : 0.0f;
            // ---- query factor (interleaved "center") ----
            const int fq = (fk / 9) * 9 + 4;
            const int pq = fq >> 12;
            const int lq = fq & 4095;
            const int hq = (lq >> 6) + pq / 3 - 1;
            const int wq = (lq & 63) + pq % 3 - 1;
            const float qv = ((unsigned)hq < 64u && (unsigned)wq < 64u)
                                 ? lds_qry[(hq << 6) + wq] : 0.0f;
            rv[j] = kv * qv;
        }
        // 302 MB write stream > 192 MB L2: non-temporal coalesced b128 stores.
        __builtin_nontemporal_store(rv, &outv[v]);
    }
}

extern "C" void kernel_launch(void* const* d_in, const int* in_sizes, int n_in,
                              void* d_out, int out_size, void* d_ws, size_t ws_size,
                              hipStream_t stream) {
    (void)n_in; (void)out_size; (void)d_ws; (void)ws_size;
    const float* key = (const float*)d_in[0];
    const float* qry = (const float*)d_in[1];
    float* out = (float*)d_out;
    const int channels = in_sizes[0] / HW;              // 8*256 = 2048
    ac_mul_kernel<<<dim3(channels), dim3(TPB), 0, stream>>>(key, qry, out);
}